// TopoSignature_42563125903699
// MI455X (gfx1250) — compile-verified
//
#include <hip/hip_runtime.h>
#include <hip/hip_bf16.h>

#define N 4096
#define NTHREADS 1024
#define EPT 4              // N / NTHREADS elements owned per thread
#define NWAVES (NTHREADS / 32)

// One block = one full Prim's MST over a dense 4096x4096 f32 distance matrix.
// block 0: tree over d1, then sum (d1[e]-d2[e])^2 over its edges
// block 1: tree over d2, then sum (d2[e]-d1[e])^2 over its edges
//
// Frontier state (min_dist, parent, in_tree) lives entirely in registers:
// thread t owns nodes 4t..4t+3. Once a node is selected its registers freeze,
// so the edge list (parent[j], weight d_self[parent[j],j]) is implicitly
// stored in registers and the cross-matrix gather happens in one parallel
// post-pass instead of on the sequential critical path.
__global__ __launch_bounds__(NTHREADS) void TopoSig_prim_mst_kernel(
    const float* __restrict__ d1, const float* __restrict__ d2,
    float* __restrict__ partial)
{
    const float* dS = (blockIdx.x == 0) ? d1 : d2;  // matrix the MST is built on
    const float* dO = (blockIdx.x == 0) ? d2 : d1;  // matrix sampled at same edges

    __shared__ __align__(16) float s_row[N];            // row d[j] via async global->LDS
    __shared__ unsigned long long s_red[2][NWAVES];     // double-buffered per-wave keys
    __shared__ float s_facc[NWAVES];

    const int tid  = threadIdx.x;
    const int lane = tid & 31;
    const int wid  = tid >> 5;
    const int base = tid * EPT;                         // owned node range

    float md[EPT];
    int   par[EPT];
    unsigned intree = 0;

#pragma unroll
    for (int e = 0; e < EPT; ++e) {
        md[e]  = dS[base + e];                          // min_dist = d[0,:]
        par[e] = 0;
    }
    if (tid == 0) intree |= 1u;                         // node 0 starts in the tree

    for (int step = 0; step < N - 1; ++step) {
        // ---- local argmin over my 4 registers; key = fbits<<32 | idx ----
        // (distances are >= 0 so the f32 bit pattern is order-preserving;
        //  low index wins ties, matching jnp.argmin)
        unsigned long long key = ~0ull;
#pragma unroll
        for (int e = 0; e < EPT; ++e) {
            unsigned long long k =
                (((unsigned long long)__float_as_uint(md[e])) << 32) |
                (unsigned)(base + e);
            bool it = (intree >> e) & 1u;
            if (!it && k < key) key = k;
        }
        // ---- wave32 butterfly min-reduce ----
#pragma unroll
        for (int off = 16; off > 0; off >>= 1) {
            unsigned long long o = __shfl_xor(key, off, 32);
            if (o < key) key = o;
        }
        if (lane == 0) s_red[step & 1][wid] = key;
        __syncthreads();                                // the only barrier per step

        // ---- every wave redundantly reduces the 32 per-wave keys:
        //      butterfly leaves the global winner in ALL lanes -> no broadcast
        unsigned long long k2 = s_red[step & 1][lane];
#pragma unroll
        for (int off = 16; off > 0; off >>= 1) {
            unsigned long long o = __shfl_xor(k2, off, 32);
            if (o < k2) k2 = o;
        }
        const int j = (int)(k2 & 0xFFFFFFFFull);

        // ---- async fetch of my 16B slice of row d[j] straight into LDS ----
        {
            unsigned ldsoff = (unsigned)(unsigned long long)(&s_row[base]);
            const float* gp = dS + (size_t)j * N + base;
            asm volatile("global_load_async_to_lds_b128 %0, %1, off"
                         :: "v"(ldsoff), "v"(gp)
                         : "memory");
        }

        // ---- owner freezes node j (its md/par now hold the edge forever) ----
        if ((j >> 2) == tid) intree |= 1u << (j & 3);

        // ---- wait for my own async chunk, relax my 4 frontier entries ----
        asm volatile("s_wait_asynccnt 0" ::: "memory");
        const float4 r = *(const float4*)(&s_row[base]);
        const float rv[EPT] = { r.x, r.y, r.z, r.w };
#pragma unroll
        for (int e = 0; e < EPT; ++e) {
            bool it = (intree >> e) & 1u;
            if (!it && rv[e] < md[e]) { md[e] = rv[e]; par[e] = j; }
        }
    }

    // ---- parallel post-pass: every node except 0 was selected exactly once;
    //      md[e] = d_self[par[e], i] (selection-time weight), par[e] = parent.
    float acc = 0.0f;
#pragma unroll
    for (int e = 0; e < EPT; ++e) {
        const int i = base + e;
        if (i != 0) {
            const float diff = md[e] - dO[(size_t)par[e] * N + i];
            acc += diff * diff;
        }
    }

    // ---- block-wide deterministic sum ----
#pragma unroll
    for (int off = 16; off > 0; off >>= 1)
        acc += __shfl_xor(acc, off, 32);
    if (lane == 0) s_facc[wid] = acc;
    __syncthreads();
    if (tid == 0) {
        float t = 0.0f;
        for (int w = 0; w < NWAVES; ++w) t += s_facc[w];
        partial[blockIdx.x] = t;
    }
}

// Deterministic 2-term combine (no float atomics -> bitwise reproducible).
__global__ void TopoSig_combine_kernel(const float* __restrict__ partial,
                                       float* __restrict__ out)
{
    if (threadIdx.x == 0 && blockIdx.x == 0)
        out[0] = partial[0] + partial[1];
}

extern "C" void kernel_launch(void* const* d_in, const int* in_sizes, int n_in,
                              void* d_out, int out_size, void* d_ws, size_t ws_size,
                              hipStream_t stream) {
    const float* d1 = (const float*)d_in[0];
    const float* d2 = (const float*)d_in[1];
    float* partial  = (float*)d_ws;   // 2 floats of scratch

    TopoSig_prim_mst_kernel<<<2, NTHREADS, 0, stream>>>(d1, d2, partial);
    TopoSig_combine_kernel<<<1, 32, 0, stream>>>(partial, (float*)d_out);
}